// OrthogonalityRegularization_7164005450196
// MI455X (gfx1250) — compile-verified
//
#include <hip/hip_runtime.h>
#include <hip/hip_bf16.h>

// CDNA5 / gfx1250, wave32.
// Gram(32x32) of squared features via V_WMMA_F32_16X16X4_F32, single HBM pass.

typedef __attribute__((ext_vector_type(2))) float v2f;
typedef __attribute__((ext_vector_type(8))) float v8f;

#define BATCH 32
#define KS 512                      // K-slice per workgroup iteration
#define LDS_STRIDE (KS + 4)         // row pad: lane stride = 4 banks -> conflict-free frag loads

__launch_bounds__(128)
__global__ void ortho_gram_wmma_kernel(const float* __restrict__ x,
                                       float* __restrict__ gram,
                                       int D, int numSlices) {
  __shared__ float lds[BATCH * LDS_STRIDE];   // x^2 tile, 32 rows x 516 floats = 66 KB

  const int tid  = threadIdx.x;
  const int lane = tid & 31;
  const int wave = tid >> 5;            // 4 waves per workgroup
  const int mi   = (wave >> 1) & 1;     // tile row  (0..1)
  const int ni   = wave & 1;            // tile col  (0..1)

  v8f acc = {};                         // 16x16 f32 accumulator (8 VGPRs)

  for (int slice = blockIdx.x; slice < numSlices; slice += gridDim.x) {
    const int k0 = slice * KS;

    // ---- stage: coalesced float4 loads, square on the fly, store to LDS ----
    for (int i = tid; i < BATCH * (KS / 4); i += blockDim.x) {
      const int b = i / (KS / 4);
      const int q = (i % (KS / 4)) << 2;       // float offset within row
      const int k = k0 + q;
      float4 v = make_float4(0.f, 0.f, 0.f, 0.f);
      if (k + 3 < D) v = *(const float4*)(x + (size_t)b * D + k);
      float* dst = &lds[b * LDS_STRIDE + q];   // 16B-aligned (stride 2064B, q*4 mult of 16)
      dst[0] = v.x * v.x;
      dst[1] = v.y * v.y;
      dst[2] = v.z * v.z;
      dst[3] = v.w * v.w;
    }
    __syncthreads();

    // ---- compute: each wave -> one 16x16 tile, K in steps of 4 ----
    // A 16x4 f32 layout: lane L holds row (L&15), K = 2*(L>>4) + {0,1}  (contiguous pair)
    // B 4x16  f32 layout: lane L holds col (L&15), K = 2*(L>>4) + {0,1} (contiguous pair)
    const float* pa = &lds[(mi * 16 + (lane & 15)) * LDS_STRIDE + ((lane >> 4) << 1)];
    const float* pb = &lds[(ni * 16 + (lane & 15)) * LDS_STRIDE + ((lane >> 4) << 1)];
    #pragma unroll 8
    for (int kk = 0; kk < KS; kk += 4) {
      v2f a  = *(const v2f*)(pa + kk);         // ds_load_b64
      v2f bm = *(const v2f*)(pb + kk);         // ds_load_b64 (same regs when mi==ni row match)
      acc = __builtin_amdgcn_wmma_f32_16x16x4_f32(
          /*neg_a=*/false, a, /*neg_b=*/false, bm,
          /*c_mod=*/(short)0, acc, /*reuse_a=*/false, /*reuse_b=*/false);
    }
    __syncthreads();
  }

  // ---- fold partial tile into global 32x32 gram (f32 atomics) ----
  // C/D layout: VGPR j -> M = j + 8*(lane>=16), N = lane&15
  const int n   = ni * 16 + (lane & 15);
  const int mhi = (lane >> 4) << 3;
  #pragma unroll
  for (int j = 0; j < 8; ++j) {
    const int m = mi * 16 + mhi + j;
    atomicAdd(&gram[m * BATCH + n], acc[j]);
  }
}

__launch_bounds__(256)
__global__ void ortho_finalize_kernel(const float* __restrict__ gram,
                                      float* __restrict__ out, int D) {
  __shared__ float red[256];
  float s = 0.f;
  for (int i = threadIdx.x; i < BATCH * BATCH; i += 256)
    if ((i >> 5) != (i & 31)) s += gram[i];   // drop diagonal
  red[threadIdx.x] = s;
  __syncthreads();
  for (int off = 128; off > 0; off >>= 1) {
    if ((int)threadIdx.x < off) red[threadIdx.x] += red[threadIdx.x + off];
    __syncthreads();
  }
  if (threadIdx.x == 0) out[0] = red[0] / (float)D;
}

extern "C" void kernel_launch(void* const* d_in, const int* in_sizes, int n_in,
                              void* d_out, int out_size, void* d_ws, size_t ws_size,
                              hipStream_t stream) {
  const float* x   = (const float*)d_in[0];
  float*       out = (float*)d_out;
  float*       gram = (float*)d_ws;          // 32*32 f32 partial-sum buffer (4 KB)

  const int total = in_sizes[0];             // 32 * 3 * 224 * 224
  const int D = total / BATCH;               // 150528
  const int numSlices = (D + KS - 1) / KS;   // 294 (exact)

  hipMemsetAsync(gram, 0, BATCH * BATCH * sizeof(float), stream);
  ortho_gram_wmma_kernel<<<numSlices, 128, 0, stream>>>(x, gram, D, numSlices);
  ortho_finalize_kernel<<<1, 256, 0, stream>>>(gram, out, D);
}